// EnhancedMSTSN_37606733644297
// MI455X (gfx1250) — compile-verified
//
#include <hip/hip_runtime.h>
#include <math.h>

// ---------------------------------------------------------------------------
// CDNA5 (gfx1250) implementation. wave32. fp32 WMMA 16x16x4 for all GEMMs.
// ---------------------------------------------------------------------------

typedef float v2f __attribute__((ext_vector_type(2)));
typedef float v8f __attribute__((ext_vector_type(8)));

__device__ __forceinline__ float gelu_exact(float x) {
    // keras 'gelu' exact: 0.5*x*(1+erf(x/sqrt(2)))
    return 0.5f * x * (1.0f + erff(x * 0.70710678118654752440f));
}

// ---------------------------------------------------------------------------
// Generic fp32 WMMA GEMM: C[M,N] = act(A[M,K] @ B[K,N] + bias[N])
// One wave computes one 16x16 C tile. M,N multiples of 16; K multiple of 4.
// A-frag: lane L holds row (L&15), K-cols {k+(L>>4)*2, +1}
// B-frag: lane L holds col (L&15), K-rows {k+(L>>4)*2, +1}
// C-frag: VGPR v -> row v+(L>>4)*8, col (L&15)
// ---------------------------------------------------------------------------
__global__ void k_gemm_wmma(const float* __restrict__ A, const float* __restrict__ B,
                            const float* __restrict__ bias, float* __restrict__ C,
                            int M, int N, int K, int act) {
    int wave = (int)((blockIdx.x * blockDim.x + threadIdx.x) >> 5);
    int lane = threadIdx.x & 31;
    int tilesN = N >> 4;
    int tiles  = (M >> 4) * tilesN;
    if (wave >= tiles) return;                 // wave-uniform
    int mt = wave / tilesN;
    int nt = wave - mt * tilesN;
    int li = lane & 15;
    int hi = lane >> 4;
    int row = (mt << 4) + li;
    int col = (nt << 4) + li;

    v8f c = {};
    for (int k = 0; k < K; k += 4) {
        int ka = k + hi * 2;
        v2f a, b;
        a.x = A[(size_t)row * K + ka];
        a.y = A[(size_t)row * K + ka + 1];
        b.x = B[(size_t)ka * N + col];
        b.y = B[(size_t)(ka + 1) * N + col];
        c = __builtin_amdgcn_wmma_f32_16x16x4_f32(false, a, false, b,
                                                  (short)0, c, false, false);
    }
    float bv = bias ? bias[col] : 0.0f;
#pragma unroll
    for (int v = 0; v < 8; ++v) {
        int r = (mt << 4) + v + hi * 8;
        float val = c[v] + bv;
        if (act) val = gelu_exact(val);
        C[(size_t)r * N + col] = val;
    }
}

// ---------------------------------------------------------------------------
// adjacency: row-normalize embeddings, then cosine-sim threshold mask
// ---------------------------------------------------------------------------
__global__ void k_norm_rows(const float* __restrict__ emb, float* __restrict__ out) {
    int r = blockIdx.x * blockDim.x + threadIdx.x;
    if (r >= 400) return;
    float ss = 0.f;
#pragma unroll
    for (int c = 0; c < 16; ++c) { float v = emb[r * 16 + c]; ss += v * v; }
    float inv = rsqrtf(fmaxf(ss, 1e-12f));
#pragma unroll
    for (int c = 0; c < 16; ++c) out[r * 16 + c] = emb[r * 16 + c] * inv;
}

__global__ void k_mask(const float* __restrict__ embn, unsigned char* __restrict__ mask) {
    int idx = blockIdx.x * blockDim.x + threadIdx.x;
    if (idx >= 400 * 400) return;
    int i = idx / 400, j = idx - i * 400;
    float d = 0.f;
#pragma unroll
    for (int c = 0; c < 16; ++c) d += embn[i * 16 + c] * embn[j * 16 + c];
    mask[idx] = (d > 0.5f) ? 1 : 0;
}

// input projection: h0[r,f] = sum_c x[r,c]*W[c,f] + b[f]   (K=3, tiny)
__global__ void k_proj(const float* __restrict__ x, const float* __restrict__ w,
                       const float* __restrict__ b, float* __restrict__ h, int rows) {
    int idx = blockIdx.x * blockDim.x + threadIdx.x;
    if (idx >= rows * 16) return;
    int r = idx >> 4, f = idx & 15;
    float acc = b[f];
#pragma unroll
    for (int c = 0; c < 3; ++c) acc += x[(size_t)r * 3 + c] * w[c * 16 + f];
    h[idx] = acc;
}

// ---------------------------------------------------------------------------
// GAT attention coefficients: es/ed[r,h] = dot(hw[r,h,:], a_src/dst[h,:])
// ---------------------------------------------------------------------------
template <int D>
__global__ void k_attn_coef(const float* __restrict__ hw, const float* __restrict__ asrc,
                            const float* __restrict__ adst, float* __restrict__ es,
                            float* __restrict__ ed, int rows) {
    int idx = blockIdx.x * blockDim.x + threadIdx.x;
    if (idx >= rows * 4) return;
    int r = idx >> 2, h = idx & 3;
    const float* hp = hw + ((size_t)r * 4 + h) * D;
    float s = 0.f, t = 0.f;
#pragma unroll
    for (int d = 0; d < D; ++d) {
        float v = hp[d];
        s += v * asrc[h * D + d];
        t += v * adst[h * D + d];
    }
    es[idx] = s;
    ed[idx] = t;
}

// ---------------------------------------------------------------------------
// GAT masked-softmax aggregation. One wave per (graph bs, node i, head h).
// out[bs,i,h*D+d] = sum_j softmax_j(leaky(ed_i+es_j), mask) * hw[bs,j,h,d] + b
// ---------------------------------------------------------------------------
template <int D, int ACT>
__global__ void k_gat_agg(const float* __restrict__ hw, const float* __restrict__ es,
                          const float* __restrict__ ed, const unsigned char* __restrict__ mask,
                          const float* __restrict__ bias, float* __restrict__ out) {
    const int F = 4 * D;
    int bs = blockIdx.x / 400;
    int i  = blockIdx.x - bs * 400;
    int h  = threadIdx.x >> 5;
    int lane = threadIdx.x & 31;

    float edi = ed[((size_t)bs * 400 + i) * 4 + h];
    const unsigned char* mrow = mask + (size_t)i * 400;
    const float* esb = es + (size_t)bs * 400 * 4;

    // pass 1: masked max (diagonal always valid -> at least one element)
    float mx = -3.0e38f;
    for (int j = lane; j < 400; j += 32) {
        float e = edi + esb[j * 4 + h];
        e = (e > 0.f) ? e : 0.2f * e;        // leaky_relu(0.2)
        e = mrow[j] ? e : -1e9f;
        mx = fmaxf(mx, e);
    }
#pragma unroll
    for (int off = 16; off > 0; off >>= 1) mx = fmaxf(mx, __shfl_xor(mx, off, 32));

    // pass 2: weighted accumulation
    float den = 0.f;
    float num[D];
#pragma unroll
    for (int d = 0; d < D; ++d) num[d] = 0.f;
    const float* hwb = hw + (size_t)bs * 400 * F;
    for (int j = lane; j < 400; j += 32) {
        float e = edi + esb[j * 4 + h];
        e = (e > 0.f) ? e : 0.2f * e;
        e = mrow[j] ? e : -1e9f;
        float w = __expf(e - mx);
        den += w;
        const float* hp = hwb + (size_t)j * F + h * D;
#pragma unroll
        for (int d = 0; d < D; ++d) num[d] += w * hp[d];
    }
#pragma unroll
    for (int off = 16; off > 0; off >>= 1) {
        den += __shfl_xor(den, off, 32);
#pragma unroll
        for (int d = 0; d < D; ++d) num[d] += __shfl_xor(num[d], off, 32);
    }
    if (lane == 0) {
        float inv = 1.0f / den;
#pragma unroll
        for (int d = 0; d < D; ++d) {
            float val = num[d] * inv + bias[h * D + d];
            if (ACT) val = gelu_exact(val);
            out[((size_t)bs * 400 + i) * F + h * D + d] = val;
        }
    }
}

// ---------------------------------------------------------------------------
// MHA core: per (sequence p, head h) wave. S=16, dk=16, DM=32, 2 heads.
// scores = QK^T/4 (WMMA) -> row softmax (shfl) -> alpha restriped via LDS ->
// O = alpha @ V (WMMA).
// Grid must be exact: 1600 seqs * 2 heads = 3200 waves = 800 blocks of 128.
// ---------------------------------------------------------------------------
__global__ void k_mha_core(const float* __restrict__ q, const float* __restrict__ k,
                           const float* __restrict__ v, float* __restrict__ o) {
    __shared__ float alpha[4][16][17];
    int w    = threadIdx.x >> 5;
    int gw   = blockIdx.x * 4 + w;
    int p    = gw >> 1;
    int h    = gw & 1;
    int lane = threadIdx.x & 31;
    int li   = lane & 15;
    int hi   = lane >> 4;

    const float* qb = q + (size_t)p * 512 + h * 16;   // [i*32 + kk]
    const float* kb = k + (size_t)p * 512 + h * 16;
    const float* vb = v + (size_t)p * 512 + h * 16;

    // s = Q @ K^T : A[i][kk]=q[i,kk], B[kk][j]=k[j,kk]
    v8f s = {};
    for (int kk = 0; kk < 16; kk += 4) {
        int ka = kk + hi * 2;
        v2f a, b;
        a.x = qb[li * 32 + ka]; a.y = qb[li * 32 + ka + 1];
        b.x = kb[li * 32 + ka]; b.y = kb[li * 32 + ka + 1];
        s = __builtin_amdgcn_wmma_f32_16x16x4_f32(false, a, false, b,
                                                  (short)0, s, false, false);
    }
    // softmax over j (the 16-lane groups), scale 1/sqrt(dk)=0.25
#pragma unroll
    for (int vv = 0; vv < 8; ++vv) {
        float val = s[vv] * 0.25f;
        float m = val;
#pragma unroll
        for (int off = 1; off < 16; off <<= 1) m = fmaxf(m, __shfl_xor(m, off, 32));
        float e = __expf(val - m);
        float sum = e;
#pragma unroll
        for (int off = 1; off < 16; off <<= 1) sum += __shfl_xor(sum, off, 32);
        alpha[w][vv + hi * 8][li] = e / sum;   // C-layout -> LDS tile [row][col]
    }
    __syncthreads();

    // O = alpha @ V : A[i][jj]=alpha[i,jj] (from LDS), B[jj][d]=v[jj,d]
    v8f oc = {};
    for (int kk = 0; kk < 16; kk += 4) {
        int ka = kk + hi * 2;
        v2f a, b;
        a.x = alpha[w][li][ka]; a.y = alpha[w][li][ka + 1];
        b.x = vb[ka * 32 + li]; b.y = vb[(ka + 1) * 32 + li];
        oc = __builtin_amdgcn_wmma_f32_16x16x4_f32(false, a, false, b,
                                                   (short)0, oc, false, false);
    }
    float* ob = o + (size_t)p * 512 + h * 16;
#pragma unroll
    for (int vv = 0; vv < 8; ++vv) ob[(vv + hi * 8) * 32 + li] = oc[vv];
}

// ---------------------------------------------------------------------------
// residual add + LayerNorm over last dim (32). One wave per row.
// ---------------------------------------------------------------------------
__global__ void k_add_ln(const float* __restrict__ x, const float* __restrict__ r,
                         const float* __restrict__ g, const float* __restrict__ b,
                         float* __restrict__ out, int rows) {
    int w = (int)((blockIdx.x * blockDim.x + threadIdx.x) >> 5);
    int lane = threadIdx.x & 31;
    if (w >= rows) return;
    float val = x[(size_t)w * 32 + lane] + r[(size_t)w * 32 + lane];
    float s = val;
#pragma unroll
    for (int off = 1; off < 32; off <<= 1) s += __shfl_xor(s, off, 32);
    float mean = s * (1.0f / 32.0f);
    float d = val - mean;
    float vs = d * d;
#pragma unroll
    for (int off = 1; off < 32; off <<= 1) vs += __shfl_xor(vs, off, 32);
    float var = vs * (1.0f / 32.0f);
    out[(size_t)w * 32 + lane] = d * rsqrtf(var + 1e-3f) * g[lane] + b[lane];
}

// ---------------------------------------------------------------------------
// mean over S=16 + MLP readout (32->16 gelu ->1). One thread per row p.
// ---------------------------------------------------------------------------
__global__ void k_readout(const float* __restrict__ t2, const float* __restrict__ w1,
                          const float* __restrict__ b1, const float* __restrict__ w2,
                          const float* __restrict__ b2, float* __restrict__ out) {
    int p = blockIdx.x * blockDim.x + threadIdx.x;
    if (p >= 1600) return;
    float m[32];
#pragma unroll
    for (int d = 0; d < 32; ++d) m[d] = 0.f;
    const float* tp = t2 + (size_t)p * 512;
    for (int s = 0; s < 16; ++s)
#pragma unroll
        for (int d = 0; d < 32; ++d) m[d] += tp[s * 32 + d];
    float acc = b2[0];
    for (int kk = 0; kk < 16; ++kk) {
        float hs = b1[kk];
#pragma unroll
        for (int d = 0; d < 32; ++d) hs += (m[d] * (1.0f / 16.0f)) * w1[d * 16 + kk];
        acc += gelu_exact(hs) * w2[kk];
    }
    out[p] = acc;
}

// ---------------------------------------------------------------------------
extern "C" void kernel_launch(void* const* d_in, const int* in_sizes, int n_in,
                              void* d_out, int out_size, void* d_ws, size_t ws_size,
                              hipStream_t stream) {
    const float* x        = (const float*)d_in[0];
    const float* adj_emb  = (const float*)d_in[1];
    const float* proj_w   = (const float*)d_in[2];
    const float* proj_b   = (const float*)d_in[3];
    const float* g1_w     = (const float*)d_in[4];
    const float* g1_asrc  = (const float*)d_in[5];
    const float* g1_adst  = (const float*)d_in[6];
    const float* g1_b     = (const float*)d_in[7];
    const float* g2_w     = (const float*)d_in[8];
    const float* g2_asrc  = (const float*)d_in[9];
    const float* g2_adst  = (const float*)d_in[10];
    const float* g2_b     = (const float*)d_in[11];
    const float* wq       = (const float*)d_in[12];
    const float* bq       = (const float*)d_in[13];
    const float* wk       = (const float*)d_in[14];
    const float* bk       = (const float*)d_in[15];
    const float* wv       = (const float*)d_in[16];
    const float* bv       = (const float*)d_in[17];
    const float* wo       = (const float*)d_in[18];
    const float* bo       = (const float*)d_in[19];
    const float* ln1_g    = (const float*)d_in[20];
    const float* ln1_b    = (const float*)d_in[21];
    const float* f_w1     = (const float*)d_in[22];
    const float* f_b1     = (const float*)d_in[23];
    const float* f_w2     = (const float*)d_in[24];
    const float* f_b2     = (const float*)d_in[25];
    const float* ln2_g    = (const float*)d_in[26];
    const float* ln2_b    = (const float*)d_in[27];
    const float* r_w1     = (const float*)d_in[28];
    const float* r_b1     = (const float*)d_in[29];
    const float* r_w2     = (const float*)d_in[30];
    const float* r_b2     = (const float*)d_in[31];

    const int ROWS = 64 * 400;        // B*S graphs, N nodes -> 25600 node rows
    const int TROW = 25600;           // 1600 sequences * 16 steps, 32-wide

    // scratch sub-allocation (256B aligned)
    char* ws = (char*)d_ws;
    size_t off = 0;
    auto alloc = [&](size_t bytes) -> void* {
        void* p = ws + off;
        off += (bytes + 255) & ~(size_t)255;
        return p;
    };
    float* embn        = (float*)alloc(400 * 16 * sizeof(float));
    unsigned char* msk = (unsigned char*)alloc(400 * 400);
    float* h0   = (float*)alloc((size_t)ROWS * 16 * sizeof(float));
    float* hw1  = (float*)alloc((size_t)ROWS * 16 * sizeof(float));
    float* es1  = (float*)alloc((size_t)ROWS * 4 * sizeof(float));
    float* ed1  = (float*)alloc((size_t)ROWS * 4 * sizeof(float));
    float* h1   = (float*)alloc((size_t)ROWS * 16 * sizeof(float));
    float* hw2  = (float*)alloc((size_t)ROWS * 32 * sizeof(float));
    float* es2  = (float*)alloc((size_t)ROWS * 4 * sizeof(float));
    float* ed2  = (float*)alloc((size_t)ROWS * 4 * sizeof(float));
    float* h2   = (float*)alloc((size_t)ROWS * 32 * sizeof(float));  // == t
    float* qb   = (float*)alloc((size_t)TROW * 32 * sizeof(float));
    float* kb   = (float*)alloc((size_t)TROW * 32 * sizeof(float));
    float* vb   = (float*)alloc((size_t)TROW * 32 * sizeof(float));
    float* ob   = (float*)alloc((size_t)TROW * 32 * sizeof(float));
    float* attn = (float*)alloc((size_t)TROW * 32 * sizeof(float));
    float* t1   = (float*)alloc((size_t)TROW * 32 * sizeof(float));
    float* f1   = (float*)alloc((size_t)TROW * 64 * sizeof(float));
    float* f2   = (float*)alloc((size_t)TROW * 32 * sizeof(float));
    float* t2   = (float*)alloc((size_t)TROW * 32 * sizeof(float));
    (void)ws_size; (void)in_sizes; (void)n_in; (void)out_size;

    auto gemm = [&](const float* A, const float* B, const float* bias, float* C,
                    int M, int N, int K, int act) {
        int waves  = (M >> 4) * (N >> 4);
        int blocks = (waves + 7) / 8;      // 8 waves (256 threads) per block
        k_gemm_wmma<<<blocks, 256, 0, stream>>>(A, B, bias, C, M, N, K, act);
    };

    // adjacency mask
    k_norm_rows<<<2, 256, 0, stream>>>(adj_emb, embn);
    k_mask<<<(400 * 400 + 255) / 256, 256, 0, stream>>>(embn, msk);

    // input projection [*,3] -> [*,16]
    k_proj<<<(ROWS * 16 + 255) / 256, 256, 0, stream>>>(x, proj_w, proj_b, h0, ROWS);

    // GAT layer 1 (heads=4, D=4), GELU epilogue on aggregate
    gemm(h0, g1_w, nullptr, hw1, ROWS, 16, 16, 0);
    k_attn_coef<4><<<(ROWS * 4 + 255) / 256, 256, 0, stream>>>(hw1, g1_asrc, g1_adst, es1, ed1, ROWS);
    k_gat_agg<4, 1><<<ROWS, 128, 0, stream>>>(hw1, es1, ed1, msk, g1_b, h1);

    // GAT layer 2 (heads=4, D=8)
    gemm(h1, g2_w, nullptr, hw2, ROWS, 32, 16, 0);
    k_attn_coef<8><<<(ROWS * 4 + 255) / 256, 256, 0, stream>>>(hw2, g2_asrc, g2_adst, es2, ed2, ROWS);
    k_gat_agg<8, 0><<<ROWS, 128, 0, stream>>>(hw2, es2, ed2, msk, g2_b, h2);

    // transformer over t = h2 reinterpreted as [1600,16,32]
    gemm(h2, wq, bq, qb, TROW, 32, 32, 0);
    gemm(h2, wk, bk, kb, TROW, 32, 32, 0);
    gemm(h2, wv, bv, vb, TROW, 32, 32, 0);
    k_mha_core<<<800, 128, 0, stream>>>(qb, kb, vb, ob);
    gemm(ob, wo, bo, attn, TROW, 32, 32, 0);
    k_add_ln<<<(TROW + 7) / 8, 256, 0, stream>>>(h2, attn, ln1_g, ln1_b, t1, TROW);
    gemm(t1, f_w1, f_b1, f1, TROW, 64, 32, 1);   // FFN1 + GELU
    gemm(f1, f_w2, f_b2, f2, TROW, 32, 64, 0);   // FFN2
    k_add_ln<<<(TROW + 7) / 8, 256, 0, stream>>>(t1, f2, ln2_g, ln2_b, t2, TROW);

    // temporal mean + readout -> [B,N] = 1600 outputs
    k_readout<<<(1600 + 255) / 256, 256, 0, stream>>>(t2, r_w1, r_b1, r_w2, r_b2,
                                                      (float*)d_out);
}